// GCN_31671088840777
// MI455X (gfx1250) — compile-verified
//
#include <hip/hip_runtime.h>

typedef __attribute__((ext_vector_type(2))) float v2f;
typedef __attribute__((ext_vector_type(8))) float v8f;
typedef int v4i __attribute__((vector_size(4 * sizeof(int))));

static constexpr int N_NODES = 100000;   // multiple of 16 (6250 row tiles)
static constexpr int N_EDGES = 1600000;
static constexpr int F_INPUT = 128;
static constexpr int HDIM    = 64;
static constexpr int LDH     = 64;       // row stride of all node-feature buffers
static constexpr int C_OUT   = 40;
static constexpr int C_PAD   = 48;       // 3 WMMA column tiles

// ---------------- gfx1250 async global->LDS helpers ----------------
__device__ __forceinline__ void async_copy16(const float* g, float* l) {
#if __has_builtin(__builtin_amdgcn_global_load_async_to_lds_b128)
  __builtin_amdgcn_global_load_async_to_lds_b128(
      (__attribute__((address_space(1))) v4i*)g,
      (__attribute__((address_space(3))) v4i*)l, 0, 0);
#else
  asm volatile("global_load_async_to_lds_b128 %0, %1, off"
               :: "v"((unsigned)(unsigned long long)l), "v"(g)
               : "memory");
#endif
}
__device__ __forceinline__ void wait_async0() {
#if __has_builtin(__builtin_amdgcn_s_wait_asynccnt)
  __builtin_amdgcn_s_wait_asynccnt(0);
#else
  asm volatile("s_wait_asynccnt 0x0" ::: "memory");
#endif
}

// ---------------- degree / symmetric normalization ----------------
__global__ void gcn_deg_init(float* deg) {
  int i = blockIdx.x * blockDim.x + threadIdx.x;
  if (i < N_NODES) deg[i] = 1.0f;                 // self-loop contribution
}
__global__ void gcn_deg_count(const int* __restrict__ dst, float* deg) {
  int e = blockIdx.x * blockDim.x + threadIdx.x;
  if (e < N_EDGES) atomicAdd(&deg[dst[e]], 1.0f);
}
__global__ void gcn_deg_rsqrt(float* deg) {
  int i = blockIdx.x * blockDim.x + threadIdx.x;
  if (i < N_NODES) deg[i] = rsqrtf(deg[i]);       // deg >= 1 always
}
__global__ void gcn_pad_w3(const float* __restrict__ W3, float* __restrict__ Wp) {
  int i = blockIdx.x * blockDim.x + threadIdx.x;
  if (i < HDIM * C_PAD) {
    int k = i / C_PAD, j = i % C_PAD;
    Wp[i] = (j < C_OUT) ? W3[k * C_OUT + j] : 0.0f;
  }
}

// ---------------- dense GEMM via V_WMMA_F32_16X16X4_F32 ----------------
// out[N_NODES x Hout] = in[N_NODES x K] @ W[K x Hout] (in has row stride K).
// The 16-row A tile is shared by all waves in the block: stage it once in LDS
// via GLOBAL_LOAD_ASYNC_TO_LDS_B128 (ASYNCcnt), then each wave reads its
// fragments with ds_load_b64. LDS row stride K+4 keeps B128 writes 16B-aligned
// and makes the fragment reads bank-conflict-free.
// One wave per 16x16 tile; blockDim.x = 32 * (Hout/16); grid.x = N_NODES/16.
template <int K, int LDW>
__global__ void gcn_gemm_wmma(const float* __restrict__ in,
                              const float* __restrict__ W,
                              float* __restrict__ out) {
  constexpr int LDA = K + 4;                  // padded LDS row stride (floats)
  __shared__ float As[16 * LDA];

  const int lane = threadIdx.x & 31;
  const int wave = threadIdx.x >> 5;
  const int m16  = lane & 15;
  const int hi   = lane >> 4;          // 0: lanes 0-15 (K 0,1)  1: lanes 16-31 (K 2,3)
  const int row0 = blockIdx.x * 16;
  const int n0   = wave * 16;

  // cooperative async stage of A tile: 16 rows x K floats, 16B chunks
  constexpr int NCH = 16 * (K / 4);
  for (int c = threadIdx.x; c < NCH; c += blockDim.x) {
    int r  = c / (K / 4);
    int c4 = c % (K / 4);
    async_copy16(in + (size_t)(row0 + r) * K + c4 * 4, &As[r * LDA + c4 * 4]);
  }
  wait_async0();        // this wave's async transfers complete
  __syncthreads();      // all waves' transfers visible

  const float* arow = &As[m16 * LDA + 2 * hi];                 // A[m][k0+2*hi ..]
  const float* bcol = W + (size_t)(2 * hi) * LDW + n0 + m16;   // B[k0+2*hi][n]

  v8f acc = {};
#pragma unroll
  for (int k0 = 0; k0 < K; k0 += 4) {
    v2f a = *(const v2f*)(arow + k0);          // ds_load_b64, conflict-free
    v2f b;
    b.x = bcol[k0 * LDW];                      // immediate-offset global loads
    b.y = bcol[k0 * LDW + LDW];
    acc = __builtin_amdgcn_wmma_f32_16x16x4_f32(false, a, false, b,
                                                (short)0, acc, false, false);
  }
  // C/D layout: VGPR r -> M=r (lanes 0-15), M=r+8 (lanes 16-31); N = lane&15.
  float* orow = out + (size_t)(row0 + 8 * hi) * LDH + n0 + m16;
#pragma unroll
  for (int r = 0; r < 8; ++r) orow[(size_t)r * LDH] = acc[r];
}

// ---------------- O = bias + self-loop term (dinv^2 * T) ----------------
__global__ void gcn_init_out(const float* __restrict__ T, float* __restrict__ O,
                             const float* __restrict__ bias,
                             const float* __restrict__ dinv,
                             int width, int bias_n) {
  int idx = blockIdx.x * blockDim.x + threadIdx.x;
  if (idx >= N_NODES * LDH) return;
  int i = idx >> 6, j = idx & 63;
  if (j < width) {
    float b  = (j < bias_n) ? bias[j] : 0.0f;
    float di = dinv[i];
    O[idx] = b + T[idx] * di * di;
  }
}

// ---------------- edge scatter-add: O[dst] += norm * T[src] ----------------
// 16 lanes per edge, NE features per lane (feature data is L2-resident: 25.6MB).
template <int NE>
__global__ void gcn_aggregate(const float* __restrict__ T, float* __restrict__ O,
                              const int* __restrict__ src, const int* __restrict__ dst,
                              const float* __restrict__ dinv) {
  long long t = (long long)blockIdx.x * blockDim.x + threadIdx.x;
  int e = (int)(t >> 4);
  int l = (int)(t & 15);
  if (e >= N_EDGES) return;
  int s = src[e], d = dst[e];
  float w = dinv[s] * dinv[d];
  const float* trow = T + (size_t)s * LDH + l;
  float*       orow = O + (size_t)d * LDH + l;
#pragma unroll
  for (int j = 0; j < NE; ++j)
    atomicAdd(&orow[j * 16], trow[j * 16] * w);   // global_atomic_add_f32, no return
}

// ---------------- leaky relu (slope 0.01), full 64-wide buffer ----------------
__global__ void gcn_leaky(float* O) {
  int idx = blockIdx.x * blockDim.x + threadIdx.x;
  if (idx < N_NODES * LDH) {
    float v = O[idx];
    O[idx] = (v > 0.0f) ? v : 0.01f * v;
  }
}

// ---------------- softmax over first C_OUT cols, one wave per node ----------------
__global__ void gcn_softmax(const float* __restrict__ O, float* __restrict__ out) {
  int gid  = blockIdx.x * blockDim.x + threadIdx.x;
  int node = gid >> 5;
  int lane = gid & 31;
  if (node >= N_NODES) return;
  const float* row = O + (size_t)node * LDH;
  float v0 = (lane < C_OUT)      ? row[lane]      : -3.4e38f;
  float v1 = (lane + 32 < C_OUT) ? row[lane + 32] : -3.4e38f;
  float m = fmaxf(v0, v1);
#pragma unroll
  for (int off = 16; off > 0; off >>= 1) m = fmaxf(m, __shfl_xor(m, off));
  float e0 = (lane < C_OUT)      ? expf(v0 - m) : 0.0f;
  float e1 = (lane + 32 < C_OUT) ? expf(v1 - m) : 0.0f;
  float sum = e0 + e1;
#pragma unroll
  for (int off = 16; off > 0; off >>= 1) sum += __shfl_xor(sum, off);
  float inv = 1.0f / sum;
  if (lane < C_OUT)      out[(size_t)node * C_OUT + lane]      = e0 * inv;
  if (lane + 32 < C_OUT) out[(size_t)node * C_OUT + lane + 32] = e1 * inv;
}

extern "C" void kernel_launch(void* const* d_in, const int* in_sizes, int n_in,
                              void* d_out, int out_size, void* d_ws, size_t ws_size,
                              hipStream_t stream) {
  const float* x   = (const float*)d_in[0];
  const int*   ei  = (const int*)d_in[1];
  const int*   src = ei;             // edge_index[0]: message sources
  const int*   dst = ei + N_EDGES;   // edge_index[1]: aggregation targets
  const float* W0  = (const float*)d_in[2];
  const float* b0  = (const float*)d_in[3];
  const float* W1  = (const float*)d_in[4];
  const float* b1  = (const float*)d_in[5];
  const float* W2  = (const float*)d_in[6];
  const float* b2  = (const float*)d_in[7];
  const float* W3  = (const float*)d_in[8];
  const float* b3  = (const float*)d_in[9];
  float* outp = (float*)d_out;

  // workspace layout (floats): T | O | dinv | Wp
  float* T    = (float*)d_ws;
  float* O    = T + (size_t)N_NODES * LDH;
  float* dinv = O + (size_t)N_NODES * LDH;
  float* Wp   = dinv + N_NODES;

  const int TB = 256;
  const int gN    = (N_NODES + TB - 1) / TB;
  const int gE    = (N_EDGES + TB - 1) / TB;
  const int gNH   = (N_NODES * LDH + TB - 1) / TB;
  const int gEdge = (int)(((long long)N_EDGES * 16 + TB - 1) / TB);
  const int gSm   = (N_NODES * 32 + TB - 1) / TB;
  const int gGemm = N_NODES / 16;   // 6250 row tiles

  // normalization: deg -> dinv (in place)
  gcn_deg_init <<<gN, TB, 0, stream>>>(dinv);
  gcn_deg_count<<<gE, TB, 0, stream>>>(dst, dinv);
  gcn_deg_rsqrt<<<gN, TB, 0, stream>>>(dinv);
  gcn_pad_w3   <<<(HDIM * C_PAD + TB - 1) / TB, TB, 0, stream>>>(W3, Wp);

  // Layer 0: x(100k x 128) @ W0(128x64)
  gcn_gemm_wmma<F_INPUT, HDIM><<<gGemm, 128, 0, stream>>>(x, W0, T);
  gcn_init_out <<<gNH, TB, 0, stream>>>(T, O, b0, dinv, HDIM, HDIM);
  gcn_aggregate<4><<<gEdge, TB, 0, stream>>>(T, O, src, dst, dinv);
  gcn_leaky    <<<gNH, TB, 0, stream>>>(O);

  // Layer 1
  gcn_gemm_wmma<HDIM, HDIM><<<gGemm, 128, 0, stream>>>(O, W1, T);
  gcn_init_out <<<gNH, TB, 0, stream>>>(T, O, b1, dinv, HDIM, HDIM);
  gcn_aggregate<4><<<gEdge, TB, 0, stream>>>(T, O, src, dst, dinv);
  gcn_leaky    <<<gNH, TB, 0, stream>>>(O);

  // Layer 2
  gcn_gemm_wmma<HDIM, HDIM><<<gGemm, 128, 0, stream>>>(O, W2, T);
  gcn_init_out <<<gNH, TB, 0, stream>>>(T, O, b2, dinv, HDIM, HDIM);
  gcn_aggregate<4><<<gEdge, TB, 0, stream>>>(T, O, src, dst, dinv);
  gcn_leaky    <<<gNH, TB, 0, stream>>>(O);

  // Layer 3 (padded to 48 cols, 3 waves per block) + softmax over 40
  gcn_gemm_wmma<HDIM, C_PAD><<<gGemm, 96, 0, stream>>>(O, Wp, T);
  gcn_init_out <<<gNH, TB, 0, stream>>>(T, O, b3, dinv, C_PAD, C_OUT);
  gcn_aggregate<3><<<gEdge, TB, 0, stream>>>(T, O, src, dst, dinv);
  gcn_softmax  <<<gSm, TB, 0, stream>>>(O, outp);
}